// DynamicLinearModel_28552942584113
// MI455X (gfx1250) — compile-verified
//
#include <hip/hip_runtime.h>
#include <math.h>

// ---------------------------------------------------------------------------
// DynamicLinearModel on MI455X (gfx1250, wave32)
//
// out[t] = theta_t + X[t]·eta + Z[t]·zeta,  theta_t = a*theta_{t-1} + Z[t-1]·gamma
// a = sigmoid(G) (constant coefficient -> blocked parallel scan).
//
// Bandwidth-bound (~270 MB vs 0.2 GFLOP -> ~11.5us floor at 23.3 TB/s).
// Kernel 1 fuses all three fp32 matvecs into ONE V_WMMA_F32_16X16X4_F32
// chain over the concatenated K=128 vector W[t] = [X[t]; Z[t]]:
//   A row 0 = [eta; zeta]  ->  D[0,n] = X·eta + Z·zeta   (yp)
//   A row 1 = [0  ; gamma] ->  D[1,n] = Z·gamma          (c)
// A fragments are constant -> staged in LDS once per block; the inner loop
// is just global_load_b64 + ds_load_b64 + wmma (no divergence).
// ---------------------------------------------------------------------------

typedef __attribute__((ext_vector_type(2))) float v2f;
typedef __attribute__((ext_vector_type(8))) float v8f;

#define XDIM      64
#define KDIM      128    // concat(X row, Z row)
#define NCHUNK    32     // KDIM / 4
#define SCAN_ROWS 4096   // rows per scan block (2^12)
#define SCAN_PT   16     // rows per thread in the scan

__device__ __forceinline__ float sigmoid_of(const float* Gp) {
  return 1.0f / (1.0f + __expf(-Gp[0]));
}

// ---------------------------------------------------------------------------
// Kernel 1: per-wave pair of 16-row tiles (32 rows), fused triple matvec.
// A layout (ISA 7.12.2, 32-bit A 16x4): lane l holds row m=l&15, K-half
// (l>>4)*2 in its two A VGPRs. B mirrors it: lane n / n+16 hold column n.
// D layout: lanes 0-15 hold M=j in VGPR j (so M=0,1 -> VGPR 0,1, lanes 0-15).
// ---------------------------------------------------------------------------
__global__ __launch_bounds__(256) void k_matvec(
    const float* __restrict__ X, const float* __restrict__ Z,
    const float* __restrict__ eta, const float* __restrict__ zeta,
    const float* __restrict__ gamma,
    float* __restrict__ yp /* = d_out */, float* __restrict__ c, int T)
{
  // A fragments for all 32 K-chunks: [chunk][lane] -> float2.  8 KB.
  __shared__ float lds_a[NCHUNK * 32 * 2];

  for (int idx = threadIdx.x; idx < NCHUNK * 32; idx += 256) {
    const int k = idx >> 5, l = idx & 31;
    const int m = l & 15, khalf = (l >> 4) * 2;
    const int col = 4 * k + khalf;            // 0..126 (always even)
    float a0 = 0.0f, a1 = 0.0f;
    if (m == 0) {                             // A row 0 = [eta ; zeta]
      a0 = (col < XDIM) ? eta[col]     : zeta[col - XDIM];
      a1 = (col < XDIM) ? eta[col + 1] : zeta[col + 1 - XDIM];
    } else if (m == 1) {                      // A row 1 = [0 ; gamma]
      a0 = (col < XDIM) ? 0.0f : gamma[col - XDIM];
      a1 = (col < XDIM) ? 0.0f : gamma[col + 1 - XDIM];
    }
    lds_a[2 * idx]     = a0;
    lds_a[2 * idx + 1] = a1;
  }
  __syncthreads();

  const int lane = threadIdx.x & 31;
  const int wave = threadIdx.x >> 5;
  const int t0   = (blockIdx.x * 8 + wave) * 32;   // 32 rows per wave
  if (t0 >= T) return;                             // wave-uniform exit

  const int m     = lane & 15;
  const int khalf = (lane >> 4) * 2;
  int rA = t0 + m;       if (rA > T - 1) rA = T - 1;   // clamp (loads only)
  int rB = t0 + 16 + m;  if (rB > T - 1) rB = T - 1;
  const float* xA = X + (long long)rA * XDIM + khalf;
  const float* zA = Z + (long long)rA * XDIM + khalf;
  const float* xB = X + (long long)rB * XDIM + khalf;
  const float* zB = Z + (long long)rB * XDIM + khalf;

  v8f acc0 = {};
  v8f acc1 = {};
#pragma unroll
  for (int k = 0; k < 16; ++k) {                   // K-chunks 0..15: X half
    const v2f av = *(const v2f*)&lds_a[2 * (k * 32 + lane)];
    const v2f bA = *(const v2f*)(xA + 4 * k);
    const v2f bB = *(const v2f*)(xB + 4 * k);
    acc0 = __builtin_amdgcn_wmma_f32_16x16x4_f32(false, av, false, bA,
                                                 (short)0, acc0, false, false);
    acc1 = __builtin_amdgcn_wmma_f32_16x16x4_f32(false, av, false, bB,
                                                 (short)0, acc1, false, false);
  }
#pragma unroll
  for (int k = 0; k < 16; ++k) {                   // K-chunks 16..31: Z half
    const v2f av = *(const v2f*)&lds_a[2 * ((k + 16) * 32 + lane)];
    const v2f bA = *(const v2f*)(zA + 4 * k);
    const v2f bB = *(const v2f*)(zB + 4 * k);
    acc0 = __builtin_amdgcn_wmma_f32_16x16x4_f32(false, av, false, bA,
                                                 (short)0, acc0, false, false);
    acc1 = __builtin_amdgcn_wmma_f32_16x16x4_f32(false, av, false, bB,
                                                 (short)0, acc1, false, false);
  }

  // D[0,n] (VGPR0) = yp, D[1,n] (VGPR1) = c, for n = lane 0..15.
  if (lane < 16) {
    const int tA = t0 + lane;
    if (tA < T) { yp[tA] = acc0[0]; c[tA] = acc0[1]; }
    const int tB = t0 + 16 + lane;
    if (tB < T) { yp[tB] = acc1[0]; c[tB] = acc1[1]; }
  }
}

// ---------------------------------------------------------------------------
// Kernel 2: per-block local scan of theta (constant coefficient a).
// u_t = (t==0) ? 0 : c[t-1].  Emits local prefixes thloc[] and block total Pb.
// ---------------------------------------------------------------------------
__global__ __launch_bounds__(256) void k_localscan(
    const float* __restrict__ c, const float* __restrict__ Gp,
    float* __restrict__ thloc, float* __restrict__ Pb, int T)
{
  __shared__ float sS[256];
  __shared__ float sPre[257];
  const float a  = sigmoid_of(Gp);
  const float a2 = a * a, a4 = a2 * a2, a8 = a4 * a4;
  const float a16 = a8 * a8;

  const int base = blockIdx.x * SCAN_ROWS + threadIdx.x * SCAN_PT;
  float u[SCAN_PT];
#pragma unroll
  for (int i = 0; i < SCAN_PT; ++i) {
    const int t = base + i;
    u[i] = (t >= 1 && t < T) ? c[t - 1] : 0.0f;
  }

  float S = 0.0f;                      // chunk scan with zero seed
#pragma unroll
  for (int i = 0; i < SCAN_PT; ++i) S = fmaf(a, S, u[i]);
  sS[threadIdx.x] = S;
  __syncthreads();

  if (threadIdx.x == 0) {              // 256-entry serial carry scan (cheap)
    float pre = 0.0f;
    for (int j = 0; j < 256; ++j) { sPre[j] = pre; pre = fmaf(a16, pre, sS[j]); }
    sPre[256] = pre;
    Pb[blockIdx.x] = pre;              // block total
  }
  __syncthreads();

  float th = sPre[threadIdx.x];        // seed = prefix of earlier chunks
#pragma unroll
  for (int i = 0; i < SCAN_PT; ++i) {
    th = fmaf(a, th, u[i]);
    const int t = base + i;
    if (t < T) thloc[t] = th;
  }
}

// ---------------------------------------------------------------------------
// Kernel 3: scan the ~123 block carries: E_b = P_b + a^4096 * E_{b-1};
// store the exclusive value E_{b-1} for each block.
// ---------------------------------------------------------------------------
__global__ void k_carryscan(const float* __restrict__ Pb,
                            const float* __restrict__ Gp,
                            float* __restrict__ Eexcl, int nblocks)
{
  if (blockIdx.x != 0 || threadIdx.x != 0) return;
  const float a = sigmoid_of(Gp);
  float aL = a;
  for (int i = 0; i < 12; ++i) aL *= aL;   // a^(2^12) = a^SCAN_ROWS
  float E = 0.0f;
  for (int b = 0; b < nblocks; ++b) { Eexcl[b] = E; E = fmaf(aL, E, Pb[b]); }
}

// ---------------------------------------------------------------------------
// Kernel 4: out[t] += thloc[t] + a^(off+1) * E_{b-1}  (pow by squaring).
// ---------------------------------------------------------------------------
__global__ __launch_bounds__(256) void k_combine(
    float* __restrict__ out, const float* __restrict__ thloc,
    const float* __restrict__ Eexcl, const float* __restrict__ Gp, int T)
{
  const int t = blockIdx.x * 256 + threadIdx.x;
  if (t >= T) return;
  const float a = sigmoid_of(Gp);
  const int  b = t >> 12;                       // t / SCAN_ROWS
  int e = (t & (SCAN_ROWS - 1)) + 1;            // exponent in [1, 4096]
  float p = 1.0f, base = a;
#pragma unroll
  for (int i = 0; i < 13; ++i) { if (e & 1) p *= base; base *= base; e >>= 1; }
  out[t] += thloc[t] + p * Eexcl[b];
}

// ---------------------------------------------------------------------------
// inputs: X_t[T*64], Z_t[T*64], G[1], eta[64], zeta[64], gamma[64]
// workspace: c[T] | thloc[T] | Pb[nb] | Eexcl[nb]  (~4 MB)
// ---------------------------------------------------------------------------
extern "C" void kernel_launch(void* const* d_in, const int* in_sizes, int n_in,
                              void* d_out, int out_size, void* d_ws, size_t ws_size,
                              hipStream_t stream) {
  const float* X     = (const float*)d_in[0];
  const float* Z     = (const float*)d_in[1];
  const float* G     = (const float*)d_in[2];
  const float* eta   = (const float*)d_in[3];
  const float* zeta  = (const float*)d_in[4];
  const float* gamma = (const float*)d_in[5];
  float* out = (float*)d_out;

  const int T  = in_sizes[0] / XDIM;
  const int nb = (T + SCAN_ROWS - 1) / SCAN_ROWS;

  float* ws    = (float*)d_ws;
  float* c     = ws;                    // [T]
  float* thloc = ws + (size_t)T;        // [T]
  float* Pb    = ws + 2 * (size_t)T;    // [nb]
  float* Eexcl = Pb + nb;               // [nb]

  const int g1 = (T + 255) / 256;       // 8 waves * 32 rows per block
  k_matvec<<<g1, 256, 0, stream>>>(X, Z, eta, zeta, gamma, out, c, T);
  k_localscan<<<nb, 256, 0, stream>>>(c, G, thloc, Pb, T);
  k_carryscan<<<1, 32, 0, stream>>>(Pb, G, Eexcl, nb);
  k_combine<<<(T + 255) / 256, 256, 0, stream>>>(out, thloc, Eexcl, G, T);
}